// LongConvolution2D_53652731462255
// MI455X (gfx1250) — compile-verified
//
#include <hip/hip_runtime.h>
#include <math.h>

// ---------------------------------------------------------------------------
// LongConvolution2D on MI455X (gfx1250): spectral conv as DFT-by-GEMM on
// V_WMMA_F32_16X16X4_F32, with LDS-staged 64x64 macro-tiles filled via
// global_load_async_to_lds_b128 (ASYNCcnt) and 2x2 WMMA subtiles per wave.
// ---------------------------------------------------------------------------

typedef __attribute__((ext_vector_type(2))) float v2f;
typedef __attribute__((ext_vector_type(8))) float v8f;

#ifndef USE_ASYNC_LDS
#define USE_ASYNC_LDS 1   // flip to 0 if the async mnemonic is rejected
#endif

#define KB   32   // K-block staged per barrier
#define ASTR 36   // LDS row stride (floats) for A panels (64 x KB); 144B, 16B-mult
#define BSTR 72   // LDS row stride (floats) for B panels (KB x 64); 288B, 16B-mult

__device__ __forceinline__ v8f wmma4(v2f a, v2f b, v8f c) {
  // (neg_a, A, neg_b, B, c_mod, C, reuse_a, reuse_b)
  return __builtin_amdgcn_wmma_f32_16x16x4_f32(false, a, false, b, (short)0, c,
                                               false, false);
}

// Async global->LDS 16B copy (per-lane). the LDS address is given as a VGPR offset;
// generic pointers to __shared__ carry the LDS offset in their low 32 bits.
__device__ __forceinline__ void cp_g2s_b128(float* smem, const float* gmem) {
#if USE_ASYNC_LDS
  unsigned lds_off = (unsigned)(size_t)smem;
  asm volatile("global_load_async_to_lds_b128 %0, %1, off"
               :: "v"(lds_off), "v"(gmem) : "memory");
#else
  *(float4*)smem = *(const float4*)gmem;
#endif
}
__device__ __forceinline__ void cp_wait() {
#if USE_ASYNC_LDS
  asm volatile("s_wait_asynccnt 0x0" ::: "memory");
#endif
}

// C/D 16x16 fp32 layout: vgpr i: row = row0 + i + 8*half, col = col0 + r
__device__ __forceinline__ void store_c(float* __restrict__ D, int ldd,
                                        int row0, int col0, int r, int hf,
                                        v8f c) {
#pragma unroll
  for (int i = 0; i < 8; ++i)
    D[(size_t)(row0 + i + 8 * hf) * ldd + (col0 + r)] = c[i];
}
__device__ __forceinline__ v8f load_c(const float* __restrict__ S, int lds_,
                                      int row0, int col0, int r, int hf) {
  v8f c;
#pragma unroll
  for (int i = 0; i < 8; ++i)
    c[i] = S[(size_t)(row0 + i + 8 * hf) * lds_ + (col0 + r)];
  return c;
}

// ------------------------- twiddle table: W = e^{-2*pi*i*j*k/512} ----------
__global__ void k_twiddle(float* __restrict__ wre, float* __restrict__ wim) {
  int idx = blockIdx.x * blockDim.x + threadIdx.x;  // 0 .. 512*512-1
  int j = idx >> 9, k = idx & 511;
  int m = (j * k) & 511;  // exact
  float th = (float)m * (-2.0f * 3.14159265358979323846f / 512.0f);
  float s, c;
  sincosf(th, &s, &c);
  wre[idx] = c;
  wim[idx] = s;  // sin(-theta) = -sin(theta)
}

// ---------------------------------------------------------------------------
// Unified blocked GEMM. 64x64 macro-tile / block (4 waves, 2x2; each wave a
// 32x32 patch = 2x2 WMMA subtiles).
// MODE 0: C0 = A0*B0, C1 = A1*B0                      (column DFT: P,Q)
// MODE 1: C0 = A0*B0 - A1*B1 ; C1 = A0*B1 + A1*B0     (row DFT -> KF store)
// MODE 2: same as 1, epilogue Hadamard with (E0,E1)   (row DFT .* KF -> Z)
// MODE 3: C0 = A0*B0 + A1*B1 ; C1 = A1*B0 - A0*B1     (inverse row -> G)
// MODE 4: C0 = A0*B0 + A1*B1, epilogue + E0           (inverse col + x -> y)
// ---------------------------------------------------------------------------
template <int MODE, int NT64, int KT, int LDA, int LDB, int LDC>
__global__ void __launch_bounds__(128) k_gemm(
    const float* __restrict__ A0, const float* __restrict__ A1,
    const float* __restrict__ B0, const float* __restrict__ B1,
    const float* __restrict__ E0, const float* __restrict__ E1,
    float* __restrict__ D0, float* __restrict__ D1,
    size_t sA, size_t sB, size_t sE, size_t sC) {
  __shared__ __align__(16) float As0[64 * ASTR];
  __shared__ __align__(16) float As1[64 * ASTR];
  __shared__ __align__(16) float Bs0[KB * BSTR];
  __shared__ __align__(16) float Bs1[KB * BSTR];

  const int tid = threadIdx.x;
  const int wave = tid >> 5, lane = tid & 31;
  const int r = lane & 15, hf = lane >> 4;
  const int wr = wave >> 1, wc = wave & 1;           // 2x2 wave grid
  const int rowBase = (blockIdx.x / NT64) * 64;
  const int colBase = (blockIdx.x % NT64) * 64;
  const int img = blockIdx.y;

  const float* a0g = A0 + (size_t)img * sA;
  const float* a1g = A1 + (size_t)img * sA;
  const float* b0g = B0 + (size_t)img * sB;
  const float* b1g = nullptr;
  if (MODE != 0) b1g = B1 + (size_t)img * sB;

  v8f C0[2][2] = {}, C1[2][2] = {};

  for (int k0 = 0; k0 < KT; k0 += KB) {
    // ---- stage A panels (64 x KB, both matrices), coalesced b128 ----------
#pragma unroll
    for (int it = 0; it < 4; ++it) {
      int c = tid + it * 128;                 // 512 chunks of 4 floats
      int arow = c >> 3, akq = (c & 7) * 4;
      size_t go = (size_t)(rowBase + arow) * LDA + (size_t)(k0 + akq);
      cp_g2s_b128(&As0[arow * ASTR + akq], a0g + go);
      cp_g2s_b128(&As1[arow * ASTR + akq], a1g + go);
    }
    // ---- stage B panels (KB x 64) -----------------------------------------
#pragma unroll
    for (int it = 0; it < 4; ++it) {
      int c = tid + it * 128;
      int brow = c >> 4, bcq = (c & 15) * 4;
      size_t go = (size_t)(k0 + brow) * LDB + (size_t)(colBase + bcq);
      cp_g2s_b128(&Bs0[brow * BSTR + bcq], b0g + go);
      if (MODE != 0) cp_g2s_b128(&Bs1[brow * BSTR + bcq], b1g + go);
    }
    cp_wait();
    __syncthreads();

    // ---- compute: 8 K-steps of 4 from LDS ---------------------------------
#pragma unroll
    for (int kk = 0; kk < KB; kk += 4) {
      v2f af0[2], af1[2], bf0[2], bf1[2], bn1[2];
#pragma unroll
      for (int i = 0; i < 2; ++i) {
        const float* p = &As0[(wr * 32 + i * 16 + r) * ASTR + kk + 2 * hf];
        af0[i].x = p[0]; af0[i].y = p[1];
        const float* q = &As1[(wr * 32 + i * 16 + r) * ASTR + kk + 2 * hf];
        af1[i].x = q[0]; af1[i].y = q[1];
      }
#pragma unroll
      for (int j = 0; j < 2; ++j) {
        const float* p = &Bs0[(kk + 2 * hf) * BSTR + wc * 32 + j * 16 + r];
        bf0[j].x = p[0]; bf0[j].y = p[BSTR];
        if (MODE != 0) {
          const float* q = &Bs1[(kk + 2 * hf) * BSTR + wc * 32 + j * 16 + r];
          bf1[j].x = q[0]; bf1[j].y = q[BSTR];
        }
        if (MODE == 1 || MODE == 2 || MODE == 3) {
          bn1[j].x = -bf1[j].x; bn1[j].y = -bf1[j].y;
        }
      }
#pragma unroll
      for (int i = 0; i < 2; ++i)
#pragma unroll
        for (int j = 0; j < 2; ++j) {
          if (MODE == 0) {
            C0[i][j] = wmma4(af0[i], bf0[j], C0[i][j]);
            C1[i][j] = wmma4(af1[i], bf0[j], C1[i][j]);
          } else if (MODE == 1 || MODE == 2) {
            C0[i][j] = wmma4(af0[i], bf0[j], C0[i][j]);
            C0[i][j] = wmma4(af1[i], bn1[j], C0[i][j]);
            C1[i][j] = wmma4(af0[i], bf1[j], C1[i][j]);
            C1[i][j] = wmma4(af1[i], bf0[j], C1[i][j]);
          } else if (MODE == 3) {
            C0[i][j] = wmma4(af0[i], bf0[j], C0[i][j]);
            C0[i][j] = wmma4(af1[i], bf1[j], C0[i][j]);
            C1[i][j] = wmma4(af1[i], bf0[j], C1[i][j]);
            C1[i][j] = wmma4(af0[i], bn1[j], C1[i][j]);
          } else {  // MODE 4
            C0[i][j] = wmma4(af0[i], bf0[j], C0[i][j]);
            C0[i][j] = wmma4(af1[i], bf1[j], C0[i][j]);
          }
        }
    }
    __syncthreads();
  }

  // ---- epilogue -----------------------------------------------------------
#pragma unroll
  for (int i = 0; i < 2; ++i)
#pragma unroll
    for (int j = 0; j < 2; ++j) {
      const int rr = rowBase + wr * 32 + i * 16;
      const int cc = colBase + wc * 32 + j * 16;
      if (MODE == 2) {
        v8f kr = load_c(E0 + (size_t)img * sE, LDC, rr, cc, r, hf);
        v8f ki = load_c(E1 + (size_t)img * sE, LDC, rr, cc, r, hf);
        v8f zr, zi;
#pragma unroll
        for (int e = 0; e < 8; ++e) {
          zr[e] = C0[i][j][e] * kr[e] - C1[i][j][e] * ki[e];
          zi[e] = C0[i][j][e] * ki[e] + C1[i][j][e] * kr[e];
        }
        store_c(D0 + (size_t)img * sC, LDC, rr, cc, r, hf, zr);
        store_c(D1 + (size_t)img * sC, LDC, rr, cc, r, hf, zi);
      } else if (MODE == 4) {
        v8f xr = load_c(E0 + (size_t)img * sE, LDC, rr, cc, r, hf);
        v8f o;
#pragma unroll
        for (int e = 0; e < 8; ++e) o[e] = C0[i][j][e] + xr[e];
        store_c(D0 + (size_t)img * sC, LDC, rr, cc, r, hf, o);
      } else {
        store_c(D0 + (size_t)img * sC, LDC, rr, cc, r, hf, C0[i][j]);
        store_c(D1 + (size_t)img * sC, LDC, rr, cc, r, hf, C1[i][j]);
      }
    }
}

// ---------------------------------------------------------------------------
extern "C" void kernel_launch(void* const* d_in, const int* in_sizes, int n_in,
                              void* d_out, int out_size, void* d_ws,
                              size_t ws_size, hipStream_t stream) {
  (void)in_sizes; (void)n_in; (void)out_size; (void)ws_size;
  const float* x    = (const float*)d_in[0];  // [8][64][256][256]
  const float* filt = (const float*)d_in[1];  // [64][256][256]
  float* y = (float*)d_out;

  const size_t NW = 512 * 512, NH = 512 * 256, NI = 256 * 256;
  const int CHUNK = 32;                       // images per pipeline pass
  float* ws   = (float*)d_ws;
  float* wre  = ws;                           //  1 MB
  float* wim  = wre + NW;                     //  1 MB
  float* kfre = wim + NW;                     // 64 MB (64 channels)
  float* kfim = kfre + (size_t)64 * NW;       // 64 MB
  float* Pb   = kfim + (size_t)64 * NW;       // 16 MB (CHUNK imgs)
  float* Qb   = Pb + (size_t)CHUNK * NH;      // 16 MB
  float* zre  = Qb + (size_t)CHUNK * NH;      // 32 MB
  float* zim  = zre + (size_t)CHUNK * NW;     // 32 MB
  float* gre  = zim + (size_t)CHUNK * NW;     // 16 MB
  float* gim  = gre + (size_t)CHUNK * NH;     // 16 MB  (~258 MB total)

  // 0) twiddle tables (d_ws is not persistent across calls)
  k_twiddle<<<dim3((512 * 512) / 256), 256, 0, stream>>>(wre, wim);

  // 1) filter spectrum KF
  for (int cc = 0; cc < 64; cc += CHUNK) {
    k_gemm<0, 4, 256, 512, 256, 256><<<dim3(32, CHUNK), 128, 0, stream>>>(
        wre, wim, filt + (size_t)cc * NI, nullptr, nullptr, nullptr,
        Pb, Qb, 0, NI, 0, NH);
    k_gemm<1, 8, 256, 256, 512, 512><<<dim3(64, CHUNK), 128, 0, stream>>>(
        Pb, Qb, wre, wim, nullptr, nullptr,
        kfre + (size_t)cc * NW, kfim + (size_t)cc * NW, NH, 0, 0, NW);
  }

  // 2) per (batch, channel-chunk): forward, fused Hadamard, inverse, +x
  for (int b = 0; b < 8; ++b) {
    for (int cc = 0; cc < 64; cc += CHUNK) {
      const float* xb = x + ((size_t)b * 64 + cc) * NI;
      float*       yb = y + ((size_t)b * 64 + cc) * NI;
      k_gemm<0, 4, 256, 512, 256, 256><<<dim3(32, CHUNK), 128, 0, stream>>>(
          wre, wim, xb, nullptr, nullptr, nullptr, Pb, Qb, 0, NI, 0, NH);
      k_gemm<2, 8, 256, 256, 512, 512><<<dim3(64, CHUNK), 128, 0, stream>>>(
          Pb, Qb, wre, wim, kfre + (size_t)cc * NW, kfim + (size_t)cc * NW,
          zre, zim, NH, 0, NW, NW);
      k_gemm<3, 4, 512, 512, 512, 256><<<dim3(32, CHUNK), 128, 0, stream>>>(
          zre, zim, wre, wim, nullptr, nullptr, gre, gim, NW, 0, 0, NH);
      k_gemm<4, 4, 512, 512, 256, 256><<<dim3(16, CHUNK), 128, 0, stream>>>(
          wre, wim, gre, gim, xb, nullptr, yb, nullptr, 0, NH, NI, NI);
    }
  }
}